// SemanticLoss_36455682408585
// MI455X (gfx1250) — compile-verified
//
#include <hip/hip_runtime.h>
#include <hip/hip_bf16.h>

// SemanticLoss on MI455X (gfx1250, wave32).
// Memory-bound: 163.8 MB single read of pred -> ~7us floor at 23.3 TB/s.
// Strategy: 1 block per row, row held in registers across max/sum/log passes.
// Final scalar reduction uses V_WMMA_F32_16X16X4_F32 for the lane fold.

typedef __attribute__((ext_vector_type(2))) float v2f;
typedef __attribute__((ext_vector_type(8))) float v8f;

#define THREADS 256
#define WAVES   (THREADS / 32)
#define VPT4    5                 // float4 chunks per thread -> supports C <= 5120

__device__ __forceinline__ float wave_max32(float v) {
#pragma unroll
  for (int off = 16; off >= 1; off >>= 1)
    v = fmaxf(v, __shfl_xor(v, off, 32));
  return v;
}

__device__ __forceinline__ float wave_sum32(float v) {
#pragma unroll
  for (int off = 16; off >= 1; off >>= 1)
    v += __shfl_xor(v, off, 32);
  return v;
}

__global__ __launch_bounds__(THREADS)
void semloss_row_kernel(const float* __restrict__ pred,
                        const int*   __restrict__ tar,
                        float*       __restrict__ rowp,
                        int C) {
  const int row  = blockIdx.x;
  const int tid  = threadIdx.x;
  const int lane = tid & 31;
  const int wid  = tid >> 5;

  const float* rp  = pred + (size_t)row * (size_t)C;
  const int    tgt = tar[row];              // uniform -> scalar load

  __shared__ float red[WAVES];

  // ---- single coalesced global read of the row into registers (b128) ----
  float x[4 * VPT4];
#pragma unroll
  for (int k = 0; k < VPT4; ++k) {
    const int c4 = tid + k * THREADS;       // float4 index
    const int c  = c4 << 2;
    if (c + 3 < C) {
      const float4 v = ((const float4*)rp)[c4];
      x[4*k + 0] = v.x; x[4*k + 1] = v.y; x[4*k + 2] = v.z; x[4*k + 3] = v.w;
    } else {
#pragma unroll
      for (int u = 0; u < 4; ++u)
        x[4*k + u] = (c + u < C) ? rp[c + u] : -INFINITY;   // -inf => zero contribution
    }
  }

  // ---- pass 1: row max ----
  float m = -INFINITY;
#pragma unroll
  for (int i = 0; i < 4 * VPT4; ++i) m = fmaxf(m, x[i]);
  m = wave_max32(m);
  if (lane == 0) red[wid] = m;
  __syncthreads();
  float bm = red[0];
#pragma unroll
  for (int w = 1; w < WAVES; ++w) bm = fmaxf(bm, red[w]);
  __syncthreads();

  // ---- pass 2: exp (in place) + sum ----
  float s = 0.0f;
#pragma unroll
  for (int i = 0; i < 4 * VPT4; ++i) {
    x[i] = __expf(x[i] - bm);               // exp(-inf - bm) == 0 for padding
    s += x[i];
  }
  s = wave_sum32(s);
  if (lane == 0) red[wid] = s;
  __syncthreads();
  float bs = red[0];
#pragma unroll
  for (int w = 1; w < WAVES; ++w) bs += red[w];
  __syncthreads();

  // ---- pass 3: log-domain product over classes ----
  const float inv = 1.0f / bs;
  float ls = 0.0f;
#pragma unroll
  for (int k = 0; k < VPT4; ++k) {
#pragma unroll
    for (int u = 0; u < 4; ++u) {
      const int   c = ((tid + k * THREADS) << 2) + u;
      const float p = x[4*k + u] * inv;     // 0 for padding
      // target class contributes log(p); others log(1-p) (matches ref's f32 1-p rounding)
      ls += (c == tgt) ? __logf(p) : __logf(1.0f - p);
    }
  }
  ls = wave_sum32(ls);
  if (lane == 0) red[wid] = ls;
  __syncthreads();
  if (tid == 0) {
    float bls = red[0];
#pragma unroll
    for (int w = 1; w < WAVES; ++w) bls += red[w];
    rowp[row] = __expf(bls);                // per-row product of masked probs
  }
}

// Single-wave final reduction: 8192 row products -> scalar.
// EXEC is all ones for the whole wave, so the WMMA restriction is satisfied.
__global__ __launch_bounds__(32)
void semloss_final_kernel(const float* __restrict__ rowp,
                          float*       __restrict__ out,
                          int N) {
  const int lane = threadIdx.x;
  float s = 0.0f;
  for (int i = lane; i < N; i += 32) s += rowp[i];

  float t;
#if __has_builtin(__builtin_amdgcn_wmma_f32_16x16x4_f32)
  // Fold 32 lane partials with the matrix unit:
  // A[16x4]: vgpr0 = lane value (K0 from lanes 0-15, K2 from lanes 16-31), vgpr1 = 0.
  // B[4x16] = ones  =>  D[m][n] = s_m + s_{m+16}, broadcast over n.
  v2f a; a.x = s;   a.y = 0.0f;
  v2f b; b.x = 1.0f; b.y = 1.0f;
  v8f c = {};
  v8f d = __builtin_amdgcn_wmma_f32_16x16x4_f32(
      /*neg_a=*/false, a, /*neg_b=*/false, b,
      /*c_mod=*/(short)0, c, /*reuse_a=*/false, /*reuse_b=*/false);
  // lanes 0-15 hold rows M=r..; summing the 8 D vgprs gives s_0..s_7 + s_16..s_23 on
  // lanes 0-15 and s_8..s_15 + s_24..s_31 on lanes 16-31 (broadcast across N).
  t = d[0] + d[1] + d[2] + d[3] + d[4] + d[5] + d[6] + d[7];
  t += __shfl_xor(t, 16, 32);
#else
  t = wave_sum32(s);
#endif

  if (lane == 0) {
    const float mean = t / (float)N;
    out[0] = 0.01f * __logf(mean + 1e-12f);
  }
}

extern "C" void kernel_launch(void* const* d_in, const int* in_sizes, int n_in,
                              void* d_out, int out_size, void* d_ws, size_t ws_size,
                              hipStream_t stream) {
  const float* pred = (const float*)d_in[0];
  const int*   tar  = (const int*)d_in[1];

  const int N = in_sizes[1];                 // 16*512 rows
  const int C = in_sizes[0] / N;             // 5000 classes

  float* rowp = (float*)d_ws;                // N floats of scratch

  semloss_row_kernel<<<N, THREADS, 0, stream>>>(pred, tar, rowp, C);
  semloss_final_kernel<<<1, 32, 0, stream>>>(rowp, (float*)d_out, N);
}